// RenamedMLP_46153718563335
// MI455X (gfx1250) — compile-verified
//
#include <hip/hip_runtime.h>

typedef __attribute__((ext_vector_type(2))) float v2f;
typedef __attribute__((ext_vector_type(8))) float v8f;

#define WAVES_PER_BLOCK 8
#define TILES_PER_WAVE  4
#define ROW_STRIDE      20                    // 17 data + 3 zero pad, keeps 8B align
#define STAGE_FLOATS    (16 * ROW_STRIDE)     // x staging tile (async target)
#define ACT_FLOATS      (16 * ROW_STRIDE)     // activation round-trip tile
#define WAVE_FLOATS     (STAGE_FLOATS + ACT_FLOATS)
#define TILE_BYTES      1088L                 // 16 rows * 17 floats * 4B

__device__ __forceinline__ long lmin(long a, long b) { return a < b ? a : b; }
__device__ __forceinline__ long lmax(long a, long b) { return a > b ? a : b; }

// GLOBAL_LOAD_ASYNC_TO_LDS_B32: VDST = per-lane LDS byte address, VADDR = 64-bit
// global address, SADDR = off(NULL). IOFFSET is added to BOTH addresses.
template <int OFF>
__device__ __forceinline__ void async_load_b32(unsigned lds_addr, const float* gaddr) {
    asm volatile("global_load_async_to_lds_b32 %0, %1, off offset:%2"
                 :: "v"(lds_addr), "v"(gaddr), "n"(OFF) : "memory");
}
__device__ __forceinline__ void async_load_b32_0(unsigned lds_addr, const float* gaddr) {
    asm volatile("global_load_async_to_lds_b32 %0, %1, off"
                 :: "v"(lds_addr), "v"(gaddr) : "memory");
}
__device__ __forceinline__ void wait_asynccnt0() {
    asm volatile("s_wait_asynccnt 0x0" ::: "memory");
}
__device__ __forceinline__ void wait_dscnt0() {
    asm volatile("s_wait_dscnt 0x0" ::: "memory");
}

__device__ __forceinline__ float relu(float v) {           // single v_med3_f32
    return __builtin_amdgcn_fmed3f(v, 0.0f, 3.40282347e+38f);
}

// issue the 9 async loads for one 16x17 tile starting at byte offset tileByte
__device__ __forceinline__ void prefetch_tile(const float* __restrict__ x,
                                              long tileByte, long xBytes,
                                              const unsigned adj[9], int lane) {
    // shared base, clamped so base+0..1083 is always in range (exact for full tiles)
    long sb = lmax(0, lmin(tileByte, xBytes - TILE_BYTES));
    const float* g = (const float*)((const char*)x + sb) + lane;   // + lane*4B
    async_load_b32<  0>(adj[0], g);
    async_load_b32<128>(adj[1], g);
    async_load_b32<256>(adj[2], g);
    async_load_b32<384>(adj[3], g);
    async_load_b32<512>(adj[4], g);
    async_load_b32<640>(adj[5], g);
    async_load_b32<768>(adj[6], g);
    async_load_b32<896>(adj[7], g);
    int f8 = lane + 256; if (f8 > 271) f8 = 271;   // lanes 16-31 duplicate elem 271
    const float* g8 = (const float*)((const char*)x + (sb + 4L * f8));
    async_load_b32_0(adj[8], g8);
}

__global__ __launch_bounds__(256) void mlp7_wmma_f32_kernel(
    const float* __restrict__ x,
    const float* __restrict__ w0, const float* __restrict__ b0,
    const float* __restrict__ w1, const float* __restrict__ b1,
    const float* __restrict__ w2, const float* __restrict__ b2,
    const float* __restrict__ w3, const float* __restrict__ b3,
    const float* __restrict__ w4, const float* __restrict__ b4,
    const float* __restrict__ w5, const float* __restrict__ b5,
    const float* __restrict__ w6, const float* __restrict__ b6,
    float* __restrict__ out, int nRows)
{
    __shared__ __align__(16) float sdata[WAVES_PER_BLOCK * WAVE_FLOATS];

    const int lane = threadIdx.x & 31;
    const int wave = threadIdx.x >> 5;
    const int half = lane >> 4;
    const int n    = lane & 15;
    const int m    = lane & 15;
    float* stage = &sdata[wave * WAVE_FLOATS];
    float* act   = stage + STAGE_FLOATS;

    // zero the staging tile once: pad cols 17..19 stay zero forever
    #pragma unroll
    for (int i = 0; i < 10; ++i) stage[lane + 32 * i] = 0.0f;

    // ---------------- branch-free B-fragment preload (W^T, zero padded) -----
    const int nc = n < 15 ? n : 14;
    v2f B0[5];
    #pragma unroll
    for (int t = 0; t < 5; ++t) {
        int kk = 4 * t + 2 * half;
        float e0 = w0[nc * 17 + (kk     < 17 ? kk     : 16)];
        float e1 = w0[nc * 17 + (kk + 1 < 17 ? kk + 1 : 16)];
        e0 = (n < 15 && kk     < 17) ? e0 : 0.0f;
        e1 = (n < 15 && kk + 1 < 17) ? e1 : 0.0f;
        B0[t] = (v2f){e0, e1};
    }
    const float* wmid[5] = {w1, w2, w3, w4, w5};
    v2f Bm[5][4];
    #pragma unroll
    for (int l = 0; l < 5; ++l) {
        const float* w = wmid[l];
        #pragma unroll
        for (int t = 0; t < 4; ++t) {
            int kk = 4 * t + 2 * half;
            float e0 = w[nc * 15 + (kk     < 15 ? kk     : 14)];
            float e1 = w[nc * 15 + (kk + 1 < 15 ? kk + 1 : 14)];
            e0 = (n < 15 && kk     < 15) ? e0 : 0.0f;
            e1 = (n < 15 && kk + 1 < 15) ? e1 : 0.0f;
            Bm[l][t] = (v2f){e0, e1};
        }
    }
    v2f B6[4];
    #pragma unroll
    for (int t = 0; t < 4; ++t) {
        int kk = 4 * t + 2 * half;
        float e0 = w6[kk     < 15 ? kk     : 14];
        float e1 = w6[kk + 1 < 15 ? kk + 1 : 14];
        e0 = (n == 0 && kk     < 15) ? e0 : 0.0f;
        e1 = (n == 0 && kk + 1 < 15) ? e1 : 0.0f;
        B6[t] = (v2f){e0, e1};
    }
    const float bias0 = (n < 15) ? b0[n] : 0.0f;
    const float* bmid[5] = {b1, b2, b3, b4, b5};
    float biasm[5];
    #pragma unroll
    for (int l = 0; l < 5; ++l) biasm[l] = (n < 15) ? bmid[l][nc] : 0.0f;
    const float bias6 = (n == 0) ? b6[0] : 0.0f;

    // ---- per-lane LDS scatter addresses for the 9 async loads ----
    // adj[i] is pre-biased by -128*i because IOFFSET adds to the LDS side too.
    unsigned adj[9];
    #pragma unroll
    for (int i = 0; i < 9; ++i) {
        int f = lane + 32 * i;
        if (f > 271) f = 271;
        unsigned ldsByte = (unsigned)(uintptr_t)&stage[(f / 17) * ROW_STRIDE + (f % 17)];
        adj[i] = ldsByte - 128u * i;
    }

    const int  nTiles     = (nRows + 15) >> 4;
    const int  tileStride = gridDim.x * WAVES_PER_BLOCK;
    const int  firstTile  = blockIdx.x * WAVES_PER_BLOCK + wave;
    const long xBytes     = (long)nRows * 17L * 4L;

    // prime the async pipeline (after the zero-init stores have completed)
    wait_dscnt0();
    prefetch_tile(x, (long)(firstTile < nTiles ? firstTile : 0) * TILE_BYTES,
                  xBytes, adj, lane);

    for (int tileIdx = firstTile; tileIdx < nTiles; tileIdx += tileStride) {
        const long rowBase = (long)tileIdx * 16;

        // staged x tile is in LDS once ASYNCcnt drains
        wait_asynccnt0();
        v2f a0[5];
        #pragma unroll
        for (int t = 0; t < 5; ++t)
            a0[t] = *(const v2f*)&stage[m * ROW_STRIDE + 4 * t + 2 * half];
        // fragment reads done -> safe to let the async engine overwrite stage
        wait_dscnt0();
        {
            int nt = tileIdx + tileStride;
            if (nt >= nTiles) nt = tileIdx;      // harmless refetch on last tile
            prefetch_tile(x, (long)nt * TILE_BYTES, xBytes, adj, lane);
        }

        // ---------------- layer 0 (17 -> 15) ----------------
        v8f c;
        #pragma unroll
        for (int v = 0; v < 8; ++v) c[v] = bias0;
        #pragma unroll
        for (int t = 0; t < 5; ++t)
            c = __builtin_amdgcn_wmma_f32_16x16x4_f32(false, a0[t], false, B0[t],
                                                      (short)0, c, false, false);
        #pragma unroll
        for (int v = 0; v < 8; ++v)
            act[(v + 8 * half) * ROW_STRIDE + n] = relu(c[v]);

        // ---------------- layers 1..5 (15 -> 15) ----------------
        #pragma unroll
        for (int l = 0; l < 5; ++l) {
            v2f a[4];
            #pragma unroll
            for (int t = 0; t < 4; ++t)
                a[t] = *(const v2f*)&act[m * ROW_STRIDE + 4 * t + 2 * half];
            #pragma unroll
            for (int v = 0; v < 8; ++v) c[v] = biasm[l];
            #pragma unroll
            for (int t = 0; t < 4; ++t)
                c = __builtin_amdgcn_wmma_f32_16x16x4_f32(false, a[t], false, Bm[l][t],
                                                          (short)0, c, false, false);
            #pragma unroll
            for (int v = 0; v < 8; ++v)
                act[(v + 8 * half) * ROW_STRIDE + n] = relu(c[v]);
        }

        // ---------------- layer 6 (15 -> 1, no ReLU) ----------------
        {
            v2f a[4];
            #pragma unroll
            for (int t = 0; t < 4; ++t)
                a[t] = *(const v2f*)&act[m * ROW_STRIDE + 4 * t + 2 * half];
            #pragma unroll
            for (int v = 0; v < 8; ++v) c[v] = bias6;
            #pragma unroll
            for (int t = 0; t < 4; ++t)
                c = __builtin_amdgcn_wmma_f32_16x16x4_f32(false, a[t], false, B6[t],
                                                          (short)0, c, false, false);
            // exchange col-0 results through LDS -> one coalesced 64B NT store
            if (n == 0) {
                #pragma unroll
                for (int v = 0; v < 8; ++v) act[v + 8 * half] = c[v];
            }
            long r = rowBase + lane;
            if (lane < 16 && r < (long)nRows)
                __builtin_nontemporal_store(act[lane], &out[r]);
        }
    }
}

extern "C" void kernel_launch(void* const* d_in, const int* in_sizes, int n_in,
                              void* d_out, int out_size, void* d_ws, size_t ws_size,
                              hipStream_t stream) {
    (void)n_in; (void)d_ws; (void)ws_size; (void)out_size;
    const float* xp = (const float*)d_in[0];
    const float* w0 = (const float*)d_in[1];  const float* b0 = (const float*)d_in[2];
    const float* w1 = (const float*)d_in[3];  const float* b1 = (const float*)d_in[4];
    const float* w2 = (const float*)d_in[5];  const float* b2 = (const float*)d_in[6];
    const float* w3 = (const float*)d_in[7];  const float* b3 = (const float*)d_in[8];
    const float* w4 = (const float*)d_in[9];  const float* b4 = (const float*)d_in[10];
    const float* w5 = (const float*)d_in[11]; const float* b5 = (const float*)d_in[12];
    const float* w6 = (const float*)d_in[13]; const float* b6 = (const float*)d_in[14];
    float* out = (float*)d_out;

    const int nRows  = in_sizes[0] / 17;
    const int nTiles = (nRows + 15) / 16;
    const int tilesPerBlock = WAVES_PER_BLOCK * TILES_PER_WAVE;
    int blocks = (nTiles + tilesPerBlock - 1) / tilesPerBlock;
    if (blocks < 1) blocks = 1;

    mlp7_wmma_f32_kernel<<<dim3(blocks), dim3(256), 0, stream>>>(
        xp, w0, b0, w1, b1, w2, b2, w3, b3, w4, b4, w5, b5, w6, b6, out, nRows);
}